// PointTransformerCosmo_59820304499245
// MI455X (gfx1250) — compile-verified
//
#include <hip/hip_runtime.h>
#include <math.h>

typedef __attribute__((ext_vector_type(2))) float v2f;
typedef __attribute__((ext_vector_type(8))) float v8f;

#define CCH 64
#define LNEPS 1e-5f

__device__ __forceinline__ float gelu_exact(float x) {
    return 0.5f * x * (1.0f + erff(x * 0.70710678118654752440f));
}

// Monotonic float atomic max via signed-max / unsigned-min bit trick.
__device__ __forceinline__ void atomic_max_f32(float* addr, float val) {
    if (val >= 0.0f) atomicMax((int*)addr, __float_as_int(val));
    else             atomicMin((unsigned int*)addr, __float_as_uint(val));
}

// ---------------------------------------------------------------------------
// Init: segmax = -FLT_MAX, denom = 0, num (=d_out) = 0
// ---------------------------------------------------------------------------
__global__ void ptc_init(float* __restrict__ segmax, float* __restrict__ denom,
                         float* __restrict__ num, int n) {
    int i = blockIdx.x * blockDim.x + threadIdx.x;
    if (i < n) {
        segmax[i] = -3.402823466e+38f;
        denom[i]  = 0.0f;
        num[i]    = 0.0f;
    }
}

// ---------------------------------------------------------------------------
// Node projections: w1/w2/w3 = features @ W1/W2/W3  (M x 64 @ 64 x 64)
// One wave computes a 16x64 tile for all three weights via f32 WMMA.
// wsel loop fully unrolled -> compile-time output pointer -> global stores.
// ---------------------------------------------------------------------------
__global__ __launch_bounds__(256)
void ptc_node_proj(const float* __restrict__ feat,
                   const float* __restrict__ W1, const float* __restrict__ W2,
                   const float* __restrict__ W3,
                   float* __restrict__ w1, float* __restrict__ w2,
                   float* __restrict__ w3, int M) {
    __shared__ float WS[3][CCH * CCH];   // 48 KB
    int t = threadIdx.x;
    for (int i = t; i < CCH * CCH; i += 256) {
        WS[0][i] = W1[i]; WS[1][i] = W2[i]; WS[2][i] = W3[i];
    }
    __syncthreads();

    int wave = t >> 5, lane = t & 31;
    int half = lane >> 4, ln16 = lane & 15;
    int row0 = (blockIdx.x * 8 + wave) * 16;
    if (row0 >= M) return;

    // A fragments: lane holds row = ln16, K slots k = 4*kk + 2*half + {0,1}
    float a[32];
    const float* frow = feat + (size_t)(row0 + ln16) * CCH;
#pragma unroll
    for (int kk = 0; kk < 16; ++kk) {
        int k0 = 4 * kk + 2 * half;
        a[2 * kk]     = frow[k0];
        a[2 * kk + 1] = frow[k0 + 1];
    }

#pragma unroll
    for (int wsel = 0; wsel < 3; ++wsel) {
        float* o = (wsel == 0) ? w1 : (wsel == 1) ? w2 : w3;
#pragma unroll
        for (int nt = 0; nt < 4; ++nt) {
            int col = nt * 16 + ln16;
            v8f acc = {0.f, 0.f, 0.f, 0.f, 0.f, 0.f, 0.f, 0.f};
#pragma unroll
            for (int kk = 0; kk < 16; ++kk) {
                v2f av; av.x = a[2 * kk]; av.y = a[2 * kk + 1];
                int k0 = 4 * kk + 2 * half;
                v2f bv; bv.x = WS[wsel][k0 * CCH + col];
                        bv.y = WS[wsel][(k0 + 1) * CCH + col];
                acc = __builtin_amdgcn_wmma_f32_16x16x4_f32(
                    false, av, false, bv, (short)0, acc, false, false);
            }
#pragma unroll
            for (int r = 0; r < 8; ++r) {
                int mrow = row0 + 8 * half + r;   // D layout: half selects row block
                o[(size_t)mrow * CCH + col] = acc[r];
            }
        }
    }
}

// ---------------------------------------------------------------------------
// Edge passes. Each wave owns 16 edges; recomputes the positional MLP
// (lin1 -> LN -> GELU -> WMMA lin2 -> LN -> GELU) fully in registers/LDS.
// PASS 1: atomic-max logits into segmax.
// PASS 2: e = exp(logit - segmax); atomicAdd denom and numerator.
// ---------------------------------------------------------------------------
template <int PASS>
__global__ __launch_bounds__(256)
void ptc_edge_pass(const int* __restrict__ source, const int* __restrict__ target,
                   const float* __restrict__ coords,
                   const float* __restrict__ Wd1, const float* __restrict__ bd1,
                   const float* __restrict__ g1, const float* __restrict__ b1,
                   const float* __restrict__ Wd2, const float* __restrict__ bd2,
                   const float* __restrict__ g2, const float* __restrict__ b2,
                   const float* __restrict__ w1, const float* __restrict__ w2,
                   const float* __restrict__ w3,
                   float* __restrict__ segmax, float* __restrict__ denom,
                   float* __restrict__ num, int E) {
    __shared__ float Wd2S[CCH * CCH];    // 16 KB
    __shared__ float Wd1S[3 * CCH];
    __shared__ float bd1S[CCH], g1S[CCH], b1S[CCH];
    __shared__ float bd2S[CCH], g2S[CCH], b2S[CCH];

    int t = threadIdx.x;
    for (int i = t; i < CCH * CCH; i += 256) Wd2S[i] = Wd2[i];
    for (int i = t; i < 3 * CCH; i += 256)   Wd1S[i] = Wd1[i];
    if (t < CCH) {
        bd1S[t] = bd1[t]; g1S[t] = g1[t]; b1S[t] = b1[t];
        bd2S[t] = bd2[t]; g2S[t] = g2[t]; b2S[t] = b2[t];
    }
    __syncthreads();

    int wave = t >> 5, lane = t & 31;
    int half = lane >> 4, ln16 = lane & 15;
    int ebase = (blockIdx.x * 8 + wave) * 16;
    if (ebase >= E) return;

    // ---- lin1 + LN + GELU, directly in WMMA A-fragment layout -------------
    int e0 = ebase + ln16;                 // A-row owned by this lane
    int eclamp = e0 < E ? e0 : (E - 1);
    float x0 = coords[(size_t)eclamp * 3 + 0];
    float x1 = coords[(size_t)eclamp * 3 + 1];
    float x2 = coords[(size_t)eclamp * 3 + 2];

    float h[32];
    float s = 0.f, s2 = 0.f;
#pragma unroll
    for (int q = 0; q < 16; ++q) {
#pragma unroll
        for (int j = 0; j < 2; ++j) {
            int c = 4 * q + 2 * half + j;
            float v = x0 * Wd1S[0 * CCH + c] + x1 * Wd1S[1 * CCH + c] +
                      x2 * Wd1S[2 * CCH + c] + bd1S[c];
            h[2 * q + j] = v; s += v; s2 += v * v;
        }
    }
    // partner lane (other half, same row) holds the complementary 32 channels
    s  += __shfl_xor(s, 16, 32);
    s2 += __shfl_xor(s2, 16, 32);
    float mu  = s * (1.f / 64.f);
    float var = s2 * (1.f / 64.f) - mu * mu;
    float inv = rsqrtf(var + LNEPS);
#pragma unroll
    for (int q = 0; q < 16; ++q) {
#pragma unroll
        for (int j = 0; j < 2; ++j) {
            int c = 4 * q + 2 * half + j;
            float v = (h[2 * q + j] - mu) * inv * g1S[c] + b1S[c];
            h[2 * q + j] = gelu_exact(v);
        }
    }

    // ---- lin2 via f32 WMMA: 16 edges x 64 channels -----------------------
    v8f zero = {0.f, 0.f, 0.f, 0.f, 0.f, 0.f, 0.f, 0.f};
    v8f acc[4] = {zero, zero, zero, zero};
#pragma unroll
    for (int kk = 0; kk < 16; ++kk) {
        v2f av; av.x = h[2 * kk]; av.y = h[2 * kk + 1];
        int k0 = 4 * kk + 2 * half;
#pragma unroll
        for (int nt = 0; nt < 4; ++nt) {
            int col = nt * 16 + ln16;
            v2f bv; bv.x = Wd2S[k0 * CCH + col];
                    bv.y = Wd2S[(k0 + 1) * CCH + col];
            acc[nt] = __builtin_amdgcn_wmma_f32_16x16x4_f32(
                false, av, false, bv, (short)0, acc[nt], false, false);
        }
    }

    // ---- + bd2, then LN over 64 channels per edge row, then GELU ---------
#pragma unroll
    for (int nt = 0; nt < 4; ++nt) {
        float bb = bd2S[nt * 16 + ln16];
#pragma unroll
        for (int r = 0; r < 8; ++r) acc[nt][r] += bb;
    }
#pragma unroll
    for (int r = 0; r < 8; ++r) {
        float rs = acc[0][r] + acc[1][r] + acc[2][r] + acc[3][r];
        float rq = acc[0][r] * acc[0][r] + acc[1][r] * acc[1][r] +
                   acc[2][r] * acc[2][r] + acc[3][r] * acc[3][r];
#pragma unroll
        for (int m = 1; m < 16; m <<= 1) {   // reduce within 16-lane half
            rs += __shfl_xor(rs, m, 32);
            rq += __shfl_xor(rq, m, 32);
        }
        float mu2 = rs * (1.f / 64.f);
        float vr  = rq * (1.f / 64.f) - mu2 * mu2;
        float iv  = rsqrtf(vr + LNEPS);
#pragma unroll
        for (int nt = 0; nt < 4; ++nt) {
            int col = nt * 16 + ln16;
            float v = (acc[nt][r] - mu2) * iv * g2S[col] + b2S[col];
            acc[nt][r] = gelu_exact(v);      // d[edge][col], in D layout
        }
    }

    // ---- scatter epilogue -------------------------------------------------
#pragma unroll
    for (int r = 0; r < 8; ++r) {
        int erow = ebase + 8 * half + r;     // D layout row -> edge id
        if (erow >= E) continue;
        int tgt = target[erow];
        int src = source[erow];
#pragma unroll
        for (int nt = 0; nt < 4; ++nt) {
            int col = nt * 16 + ln16;
            float dv = acc[nt][r];
            size_t ti = (size_t)tgt * CCH + col;
            size_t si = (size_t)src * CCH + col;
            float logit = w1[ti] - w2[si] + dv;
            if (PASS == 1) {
                atomic_max_f32(&segmax[ti], logit);
            } else {
                float ev = expf(logit - segmax[ti]);
                atomicAdd(&denom[ti], ev);
                atomicAdd(&num[ti], ev * (w3[si] + dv));
            }
        }
    }
}

// ---------------------------------------------------------------------------
// Finalize: out = denom>0 ? num/denom : 0  (num lives in d_out)
// ---------------------------------------------------------------------------
__global__ void ptc_finalize(float* __restrict__ out, const float* __restrict__ denom,
                             int n) {
    int i = blockIdx.x * blockDim.x + threadIdx.x;
    if (i < n) {
        float d = denom[i];
        out[i] = (d > 0.0f) ? out[i] / d : 0.0f;
    }
}

// ---------------------------------------------------------------------------
extern "C" void kernel_launch(void* const* d_in, const int* in_sizes, int n_in,
                              void* d_out, int out_size, void* d_ws, size_t ws_size,
                              hipStream_t stream) {
    const int*   source = (const int*)d_in[0];
    const int*   target = (const int*)d_in[1];
    const float* feat   = (const float*)d_in[2];
    const float* coords = (const float*)d_in[3];
    const float* Wd1    = (const float*)d_in[4];
    const float* bd1    = (const float*)d_in[5];
    const float* g1     = (const float*)d_in[6];
    const float* b1     = (const float*)d_in[7];
    const float* Wd2    = (const float*)d_in[8];
    const float* bd2    = (const float*)d_in[9];
    const float* g2     = (const float*)d_in[10];
    const float* b2     = (const float*)d_in[11];
    const float* W1     = (const float*)d_in[12];
    const float* W2     = (const float*)d_in[13];
    const float* W3     = (const float*)d_in[14];

    int E = in_sizes[0];
    int M = in_sizes[2] / CCH;
    int n = M * CCH;

    float* ws     = (float*)d_ws;
    float* w1     = ws;
    float* w2     = w1 + (size_t)n;
    float* w3     = w2 + (size_t)n;
    float* segmax = w3 + (size_t)n;
    float* denom  = segmax + (size_t)n;
    float* num    = (float*)d_out;

    ptc_init<<<(n + 255) / 256, 256, 0, stream>>>(segmax, denom, num, n);

    int nodeTiles = (M + 15) / 16;
    ptc_node_proj<<<(nodeTiles + 7) / 8, 256, 0, stream>>>(
        feat, W1, W2, W3, w1, w2, w3, M);

    int edgeTiles = (E + 15) / 16;
    int edgeBlocks = (edgeTiles + 7) / 8;
    ptc_edge_pass<1><<<edgeBlocks, 256, 0, stream>>>(
        source, target, coords, Wd1, bd1, g1, b1, Wd2, bd2, g2, b2,
        w1, w2, w3, segmax, denom, num, E);
    ptc_edge_pass<2><<<edgeBlocks, 256, 0, stream>>>(
        source, target, coords, Wd1, bd1, g1, b1, Wd2, bd2, g2, b2,
        w1, w2, w3, segmax, denom, num, E);

    ptc_finalize<<<(n + 255) / 256, 256, 0, stream>>>(num, denom, n);
}